// Model_39676907888111
// MI455X (gfx1250) — compile-verified
//
#include <hip/hip_runtime.h>
#include <hip/hip_bf16.h>

#define B_ 2
#define N_ 1024
#define M_ 1024
#define H_ 128

// 1/128^0.4 = 1/2^2.8
#define SCALE_ 0.14358729437f

typedef __attribute__((ext_vector_type(2))) float v2f;
typedef __attribute__((ext_vector_type(8))) float v8f;

// ---------------------------------------------------------------------------
// Kernel A: wcol[h] = sum_o W[o,h];  bsum = sum(b).  One block, 128 threads.
// ---------------------------------------------------------------------------
__global__ __launch_bounds__(128)
void prep_kernel(const float* __restrict__ W, const float* __restrict__ bvec,
                 float* __restrict__ wcol, float* __restrict__ bsum) {
    const int h = threadIdx.x;
    float s = 0.f;
    #pragma unroll 8
    for (int o = 0; o < H_; ++o) s += W[o * H_ + h];
    wcol[h] = s;

    __shared__ float sh[H_];
    sh[h] = bvec[h];
    __syncthreads();
    for (int off = H_ / 2; off > 0; off >>= 1) {
        if (h < off) sh[h] += sh[h + off];
        __syncthreads();
    }
    if (h == 0) *bsum = sh[0];
}

// ---------------------------------------------------------------------------
// Kernel B: p[b,m] = softmax_m( SCALE * (keys[b,m,:]·wcol + bsum) )
// One block per batch, 1024 threads = 32 waves. Each wave computes 32 rows of
// the GEMV as two 16-row V_WMMA_F32_16X16X4_F32 tile chains (B-matrix is
// wcol broadcast across all 16 columns, so every column of D = the dot
// products, computed natively in f32).
// ---------------------------------------------------------------------------
__global__ __launch_bounds__(1024)
void softmax_kernel(const float* __restrict__ keys,
                    const float* __restrict__ wcol_g,
                    const float* __restrict__ bsum_g,
                    float* __restrict__ p) {
    const int b    = blockIdx.x;
    const int tid  = threadIdx.x;
    const int wave = tid >> 5;
    const int lane = tid & 31;

    __shared__ float ks_sh[M_];
    __shared__ float wcol_sh[H_];
    __shared__ float red[32];
    __shared__ float bcast[2];

    if (tid < H_) wcol_sh[tid] = wcol_g[tid];
    __syncthreads();

#if defined(__gfx1250__) && __has_builtin(__builtin_amdgcn_wmma_f32_16x16x4_f32)
    {
        const int half = lane >> 4;   // 0: K=0,1 side   1: K=2,3 side
        const int lr   = lane & 15;   // row within tile (A-matrix M = lr)
        #pragma unroll
        for (int t = 0; t < 2; ++t) {
            const int row0 = wave * 32 + t * 16;
            const float* krow = keys + ((size_t)b * M_ + (row0 + lr)) * H_;
            v8f acc = {};
            #pragma unroll
            for (int k0 = 0; k0 < H_; k0 += 4) {
                const int kk = k0 + half * 2;
                v2f a;  a.x  = krow[kk];     a.y  = krow[kk + 1];
                v2f bb; bb.x = wcol_sh[kk];  bb.y = wcol_sh[kk + 1];
                acc = __builtin_amdgcn_wmma_f32_16x16x4_f32(
                    /*neg_a=*/false, a, /*neg_b=*/false, bb,
                    /*c_mod=*/(short)0, acc, /*reuse_a=*/false, /*reuse_b=*/false);
            }
            // D layout: VGPR j -> M=j (lanes 0-15, N=lane) / M=8+j (lanes 16-31).
            // Column N=0 of each half-wave carries the 16 dot products.
            if (lr == 0) {
                #pragma unroll
                for (int j = 0; j < 8; ++j)
                    ks_sh[row0 + half * 8 + j] = acc[j];
            }
        }
    }
#else
    {   // scalar fallback (one row per lane)
        const int row = wave * 32 + lane;
        const float* krow = keys + ((size_t)b * M_ + row) * H_;
        float acc = 0.f;
        #pragma unroll 8
        for (int k = 0; k < H_; ++k) acc += krow[k] * wcol_sh[k];
        ks_sh[row] = acc;
    }
#endif
    __syncthreads();

    const float bs    = *bsum_g;
    const float logit = SCALE_ * (ks_sh[tid] + bs);

    // block-wide max (wave32 shuffle tree, then cross-wave via LDS)
    float mx = logit;
    #pragma unroll
    for (int off = 16; off > 0; off >>= 1) mx = fmaxf(mx, __shfl_xor(mx, off, 32));
    if (lane == 0) red[wave] = mx;
    __syncthreads();
    if (wave == 0) {
        float m2 = red[lane];
        #pragma unroll
        for (int off = 16; off > 0; off >>= 1) m2 = fmaxf(m2, __shfl_xor(m2, off, 32));
        if (lane == 0) bcast[0] = m2;
    }
    __syncthreads();
    mx = bcast[0];

    const float e = __expf(logit - mx);

    // block-wide sum
    float sum = e;
    #pragma unroll
    for (int off = 16; off > 0; off >>= 1) sum += __shfl_xor(sum, off, 32);
    if (lane == 0) red[wave] = sum;
    __syncthreads();
    if (wave == 0) {
        float s2 = red[lane];
        #pragma unroll
        for (int off = 16; off > 0; off >>= 1) s2 += __shfl_xor(s2, off, 32);
        if (lane == 0) bcast[1] = s2;
    }
    __syncthreads();

    p[(size_t)b * M_ + tid] = e / bcast[1];
}

// ---------------------------------------------------------------------------
// Kernel C: out[b,n,m] = values[b,n] * p[b,m].  Pure store bandwidth (8 MB).
// One block per (b,n) row; 256 threads x float4 = 1024 floats.
// ---------------------------------------------------------------------------
__global__ __launch_bounds__(256)
void outer_kernel(const float* __restrict__ p,
                  const float* __restrict__ values,
                  float* __restrict__ out) {
    const int row = blockIdx.x;        // b*N + n
    const int b   = row >> 10;         // N_ == 1024
    const float v = values[row];       // values is [B,N,1,1] -> flat B*N

    const float4 pv = ((const float4*)(p + (size_t)b * M_))[threadIdx.x];
    float4 o;
    o.x = pv.x * v; o.y = pv.y * v; o.z = pv.z * v; o.w = pv.w * v;
    ((float4*)out)[(size_t)row * (M_ / 4) + threadIdx.x] = o;
}

extern "C" void kernel_launch(void* const* d_in, const int* in_sizes, int n_in,
                              void* d_out, int out_size, void* d_ws, size_t ws_size,
                              hipStream_t stream) {
    (void)in_sizes; (void)n_in; (void)out_size; (void)ws_size;
    // inputs: 0=queries (unused: cancels in softmax), 1=keys, 2=values, 3=W, 4=b
    const float* keys   = (const float*)d_in[1];
    const float* values = (const float*)d_in[2];
    const float* W      = (const float*)d_in[3];
    const float* bvec   = (const float*)d_in[4];
    float* out = (float*)d_out;

    float* ws   = (float*)d_ws;
    float* wcol = ws;            // 128 floats
    float* bsum = ws + 128;      // 1 float
    float* p    = ws + 160;      // 2048 floats, 16B-aligned offset (640 bytes)

    prep_kernel<<<1, 128, 0, stream>>>(W, bvec, wcol, bsum);
    softmax_kernel<<<B_, 1024, 0, stream>>>(keys, wcol, bsum, p);
    outer_kernel<<<B_ * N_, 256, 0, stream>>>(p, values, out);
}